// FCN_78443282694449
// MI455X (gfx1250) — compile-verified
//
#include <hip/hip_runtime.h>
#include <hip/hip_bf16.h>

// ---------------------------------------------------------------------------
// Problem dims (fixed by setup_inputs)
// ---------------------------------------------------------------------------
#define BATCH 128
#define TLEN  1024
#define CIN   64
#define C1    128
#define C2    256
#define C3    128
#define SPROT 16
#define NCLS  32
#define BIGV  1e30f
#define BN_EPS 1e-5f
#define COS_EPS 1e-8f

typedef float v2f __attribute__((ext_vector_type(2)));
typedef float v8f __attribute__((ext_vector_type(8)));

// ---------------------------------------------------------------------------
// Conv1d as implicit GEMM with fp32 WMMA (V_WMMA_F32_16X16X4_F32).
//
// Block = 256 threads = 8 waves, all sharing one (b, t0) 16-t window.
// The x slice [Cin x (16+KW-1)] is staged into LDS once (zero padding baked
// in), so the inner loop is ds_load (B frag) + global_load_b64 (A frag,
// L2-resident weights) + v_wmma_f32_16x16x4_f32.  Wave w handles co-tile
// (blockIdx.y*8 + w).  All tile math is compile-time (templates + 3D grid).
//
// K is split into TWO independent accumulation chains (k%8==0 and k%8==4) so
// the scheduler can interleave WMMAs instead of stalling on the D->C RAW
// hazard of a single chain; the two v8f accumulators are summed at the end.
//
// f32 A 16x4 layout: lanes 0-15 hold M=lane, VGPR0=K0,VGPR1=K1; lanes 16-31
// hold K2/K3.  B mirrors with N=lane&15.  C/D: VGPR r -> M = r + 8*(lane>=16).
// ---------------------------------------------------------------------------
template <int KW, int Cin, int Cout>
__global__ __launch_bounds__(256)
void conv_wmma_kernel(const float* __restrict__ x, const float* __restrict__ w,
                      const float* __restrict__ bias, float* __restrict__ y)
{
    constexpr int PAD = KW / 2;
    constexpr int W   = 16 + KW - 1;     // staged t-extent per channel
    constexpr int KK  = Cin * KW;        // GEMM K (multiple of 8 here)
    __shared__ float xs[Cin * W];

    const int lane = threadIdx.x & 31;
    const int wv   = threadIdx.x >> 5;   // wave in block: 0..7
    const int b    = blockIdx.z;
    const int t0   = blockIdx.x << 4;
    const int co0  = (blockIdx.y * 8 + wv) << 4;

    // ---- cooperative stage of x[b, :, t0-PAD .. t0+15+PAD] into LDS ----
    const float* __restrict__ xb = x + (size_t)b * Cin * TLEN;
    for (int i = threadIdx.x; i < Cin * W; i += 256) {
        int ci = i / W, u = i - ci * W;          // W is a constant
        int t  = t0 + u - PAD;
        xs[i] = (t >= 0 && t < TLEN) ? xb[(size_t)ci * TLEN + t] : 0.0f;
    }
    __syncthreads();

    const int hf  = lane >> 4;
    const int l15 = lane & 15;
    const float* __restrict__ wrow = w + (size_t)(co0 + l15) * KK;  // A row M=l15

    v8f acc0 = {}, acc1 = {};
    for (int k0 = 0; k0 < KK; k0 += 8) {
        __builtin_prefetch(wrow + k0 + 64, 0, 0);            // global_prefetch_b8
        // ---- chain 0: K-block [k0, k0+4) ----
        {
            const int ka = k0 + (hf << 1);       // even => 8B aligned b64 load
            v2f afrag = *(const v2f*)(wrow + ka);
            v2f bfrag;
            int ci  = ka / KW,       dk  = ka - ci * KW;     // const divisor
            int ci2 = (ka + 1) / KW, dk2 = (ka + 1) - ci2 * KW;
            bfrag.x = xs[ci * W + l15 + dk];
            bfrag.y = xs[ci2 * W + l15 + dk2];
            acc0 = __builtin_amdgcn_wmma_f32_16x16x4_f32(false, afrag, false, bfrag,
                                                         (short)0, acc0, false, false);
        }
        // ---- chain 1: K-block [k0+4, k0+8) ----
        {
            const int ka = k0 + 4 + (hf << 1);
            v2f afrag = *(const v2f*)(wrow + ka);
            v2f bfrag;
            int ci  = ka / KW,       dk  = ka - ci * KW;
            int ci2 = (ka + 1) / KW, dk2 = (ka + 1) - ci2 * KW;
            bfrag.x = xs[ci * W + l15 + dk];
            bfrag.y = xs[ci2 * W + l15 + dk2];
            acc1 = __builtin_amdgcn_wmma_f32_16x16x4_f32(false, afrag, false, bfrag,
                                                         (short)0, acc1, false, false);
        }
    }
    v8f acc = acc0 + acc1;
#pragma unroll
    for (int r = 0; r < 8; ++r) {
        const int row = r + (hf << 3);
        y[((size_t)b * Cout + co0 + row) * TLEN + t0 + l15] = acc[r] + bias[co0 + row];
    }
}

// ---------------------------------------------------------------------------
// BatchNorm training statistics per channel: mean/biased-var over (B,T);
// fold into scale = g*rsqrt(var+eps), shift = beta - mu*scale.
// ---------------------------------------------------------------------------
__global__ __launch_bounds__(256)
void bn_stats_kernel(const float* __restrict__ y, const float* __restrict__ g,
                     const float* __restrict__ beta, float* __restrict__ scale,
                     float* __restrict__ shift, int C)
{
    const int c = blockIdx.x;
    const int N = BATCH * TLEN;
    double s = 0.0, ss = 0.0;
    for (int n = threadIdx.x; n < N; n += blockDim.x) {
        int b = n >> 10, t = n & 1023;
        float v = y[((size_t)b * C + c) * TLEN + t];
        s  += v;
        ss += (double)v * v;
    }
    __shared__ double sh1[256], sh2[256];
    sh1[threadIdx.x] = s; sh2[threadIdx.x] = ss;
    __syncthreads();
    for (int st = 128; st > 0; st >>= 1) {
        if ((int)threadIdx.x < st) {
            sh1[threadIdx.x] += sh1[threadIdx.x + st];
            sh2[threadIdx.x] += sh2[threadIdx.x + st];
        }
        __syncthreads();
    }
    if (threadIdx.x == 0) {
        double mu  = sh1[0] / N;
        double var = sh2[0] / N - mu * mu;
        float inv = rsqrtf((float)var + BN_EPS);
        float sc  = g[c] * inv;
        scale[c] = sc;
        shift[c] = beta[c] - (float)mu * sc;
    }
}

__global__ __launch_bounds__(256)
void bn_relu_kernel(float* __restrict__ y, const float* __restrict__ scale,
                    const float* __restrict__ shift, int C, size_t total)
{
    size_t i = (size_t)blockIdx.x * blockDim.x + threadIdx.x;
    if (i >= total) return;
    int c = (int)((i >> 10) % C);   // T = 1024
    float v = y[i] * scale[c] + shift[c];
    y[i] = fmaxf(v, 0.0f);
}

// ---------------------------------------------------------------------------
// Prototype column norms (+EPS) : protos [C3, S] stored c*S+s
// ---------------------------------------------------------------------------
__global__ void pnorm_kernel(const float* __restrict__ protos, float* __restrict__ pnorm)
{
    int s = threadIdx.x;          // 16 threads
    float n = 0.0f;
    for (int c = 0; c < C3; ++c) { float v = protos[c * SPROT + s]; n += v * v; }
    pnorm[s] = sqrtf(n) + COS_EPS;
}

// ---------------------------------------------------------------------------
// Cosine cost matrix D[b,s,t] = 1 - <p_s, h_bt> / ((|p_s|+eps)(|h_bt|+eps))
// ---------------------------------------------------------------------------
__global__ __launch_bounds__(256)
void cost_kernel(const float* __restrict__ h, const float* __restrict__ protos,
                 const float* __restrict__ pnorm, float* __restrict__ Dm)
{
    __shared__ float psh[C3 * SPROT];
    const int b = blockIdx.y;
    const int t = blockIdx.x * blockDim.x + threadIdx.x;
    for (int i = threadIdx.x; i < C3 * SPROT; i += blockDim.x) psh[i] = protos[i];
    __syncthreads();

    float dot[SPROT];
#pragma unroll
    for (int s = 0; s < SPROT; ++s) dot[s] = 0.0f;
    float nh = 0.0f;
    for (int c = 0; c < C3; ++c) {
        float hv = h[((size_t)b * C3 + c) * TLEN + t];
        nh += hv * hv;
#pragma unroll
        for (int s = 0; s < SPROT; ++s) dot[s] += hv * psh[c * SPROT + s];
    }
    nh = sqrtf(nh) + COS_EPS;
#pragma unroll
    for (int s = 0; s < SPROT; ++s)
        Dm[((size_t)b * SPROT + s) * TLEN + t] = 1.0f - dot[s] / (pnorm[s] * nh);
}

// ---------------------------------------------------------------------------
// Soft-DTW forward (R) + exact gradient backward (E) per batch element.
// One wave (32 lanes) per b; lane L owns DP row r=L+1 (S=16 rows active).
// Anti-diagonal wavefront; neighbor rows exchanged with __shfl (wave32).
// R stored in global scratch [(S+2) x (T+2)] with -BIG boundary padding.
// gamma = 1.
// ---------------------------------------------------------------------------
__global__ __launch_bounds__(32)
void dtw_kernel(const float* __restrict__ Dm, float* __restrict__ Rbuf,
                float* __restrict__ Abuf)
{
    const int b    = blockIdx.x;
    const int lane = threadIdx.x;
    const int RT   = TLEN + 2;
    float* __restrict__ R = Rbuf + (size_t)b * (SPROT + 2) * RT;
    const float* __restrict__ D = Dm + (size_t)b * SPROT * TLEN;
    float* __restrict__ A = Abuf + (size_t)b * SPROT * TLEN;
    const int r = lane + 1;

    // boundary prefill: col T+1 and row S+1 get -BIG (matches finite-BIG ref)
    for (int i = lane; i < SPROT + 2; i += 32) R[i * RT + (TLEN + 1)] = -BIGV;
    for (int j = lane; j < TLEN + 2; j += 32) R[(SPROT + 1) * RT + j] = -BIGV;
    __threadfence_block();
    __syncthreads();

    // ---- forward DP ----
    float v1 = BIGV, v2 = BIGV;     // my R at diag d-1 / d-2
    for (int d = 2; d <= SPROT + TLEN; ++d) {
        float up = __shfl_up(v1, 1, 32);   // R[r-1][j]
        float dg = __shfl_up(v2, 1, 32);   // R[r-1][j-1]
        if (lane == 0) { up = BIGV; dg = (d == 2) ? 0.0f : BIGV; }
        const float left = v1;             // R[r][j-1]
        const int j = d - r;
        const bool act = (lane < SPROT) && (j >= 1) && (j <= TLEN);
        float rij = v1;
        if (act) {
            float m  = fminf(dg, fminf(up, left));
            float sm = m - logf(expf(-(dg - m)) + expf(-(up - m)) + expf(-(left - m)));
            rij = D[(r - 1) * TLEN + (j - 1)] + sm;   // softmin, gamma = 1
            R[r * RT + j] = rij;
        }
        v2 = v1;
        v1 = rij;
    }
    __threadfence_block();
    __syncthreads();
    if (lane == 0) R[(SPROT + 1) * RT + (TLEN + 1)] = R[SPROT * RT + TLEN];
    __threadfence_block();
    __syncthreads();

    // ---- backward (E recursion, exact dR/dD) ----
    float e1 = 0.0f, e2 = 0.0f;     // my E at diag d+1 / d+2
    for (int d = SPROT + TLEN; d >= 2; --d) {
        float e_up = __shfl_down(e1, 1, 32);   // E[r+1][j]
        float e_dg = __shfl_down(e2, 1, 32);   // E[r+1][j+1]
        const int j = d - r;
        if (lane == SPROT - 1) {               // row S reads boundary row S+1
            e_up = 0.0f;
            e_dg = (j == TLEN) ? 1.0f : 0.0f;
        }
        const bool act = (lane < SPROT) && (j >= 1) && (j <= TLEN);
        float e = e1;
        if (act) {
            float rij  = R[r * RT + j];
            float r_up = R[(r + 1) * RT + j];
            float r_lf = R[r * RT + (j + 1)];
            float r_dg = R[(r + 1) * RT + (j + 1)];
            float d_up = (r < SPROT) ? D[r * TLEN + (j - 1)] : 0.0f;
            float d_lf = (j < TLEN) ? D[(r - 1) * TLEN + j] : 0.0f;
            float d_dg = (r < SPROT && j < TLEN) ? D[r * TLEN + j] : 0.0f;
            float wa = expf(r_up - rij - d_up);
            float wb = expf(r_lf - rij - d_lf);
            float wc = expf(r_dg - rij - d_dg);
            e = wa * e_up + wb * e1 + wc * e_dg;
            A[(r - 1) * TLEN + (j - 1)] = e;
        }
        e2 = e1;
        e1 = act ? e : e1;
    }
}

// ---------------------------------------------------------------------------
// Normalize A rows over t (avg pooling op)
// ---------------------------------------------------------------------------
__global__ __launch_bounds__(256)
void arow_norm_kernel(float* __restrict__ A)
{
    float* row = A + (size_t)blockIdx.x * TLEN;
    float s = 0.0f;
    for (int t = threadIdx.x; t < TLEN; t += 256) s += row[t];
    __shared__ float sh[256];
    sh[threadIdx.x] = s;
    __syncthreads();
    for (int st = 128; st > 0; st >>= 1) {
        if ((int)threadIdx.x < st) sh[threadIdx.x] += sh[threadIdx.x + st];
        __syncthreads();
    }
    float inv = 1.0f / sh[0];
    for (int t = threadIdx.x; t < TLEN; t += 256) row[t] *= inv;
}

// ---------------------------------------------------------------------------
// pooled[b,c,s] = sum_t h[b,c,t] * A[b,s,t]
// ---------------------------------------------------------------------------
__global__ __launch_bounds__(256)
void pool_kernel(const float* __restrict__ h, const float* __restrict__ A,
                 float* __restrict__ pooled)
{
    const int c = blockIdx.x, b = blockIdx.y;
    const float* hrow = h + ((size_t)b * C3 + c) * TLEN;
    const float* Ab   = A + (size_t)b * SPROT * TLEN;
    float acc[SPROT];
#pragma unroll
    for (int s = 0; s < SPROT; ++s) acc[s] = 0.0f;
    for (int t = threadIdx.x; t < TLEN; t += 256) {
        float hv = hrow[t];
#pragma unroll
        for (int s = 0; s < SPROT; ++s) acc[s] += hv * Ab[s * TLEN + t];
    }
    __shared__ float sh[256 * SPROT];
#pragma unroll
    for (int s = 0; s < SPROT; ++s) sh[threadIdx.x * SPROT + s] = acc[s];
    __syncthreads();
    for (int st = 128; st > 0; st >>= 1) {
        if ((int)threadIdx.x < st)
#pragma unroll
            for (int s = 0; s < SPROT; ++s)
                sh[threadIdx.x * SPROT + s] += sh[(threadIdx.x + st) * SPROT + s];
        __syncthreads();
    }
    if (threadIdx.x < SPROT)
        pooled[((size_t)b * C3 + c) * SPROT + threadIdx.x] = sh[threadIdx.x];
}

// ---------------------------------------------------------------------------
// FC: out[b,nc] = pooled_flat[b,:] . fc_w[nc,:] + fc_b[nc]
// ---------------------------------------------------------------------------
__global__ __launch_bounds__(32)
void fc_kernel(const float* __restrict__ pooled, const float* __restrict__ w,
               const float* __restrict__ bias, float* __restrict__ out)
{
    const int b = blockIdx.x;
    const int nc = threadIdx.x;   // 32 threads
    const float* p  = pooled + (size_t)b * (C3 * SPROT);
    const float* wr = w + (size_t)nc * (C3 * SPROT);
    float acc = bias[nc];
    for (int i = 0; i < C3 * SPROT; ++i) acc += p[i] * wr[i];
    out[(size_t)b * NCLS + nc] = acc;
}

// ---------------------------------------------------------------------------
extern "C" void kernel_launch(void* const* d_in, const int* in_sizes, int n_in,
                              void* d_out, int out_size, void* d_ws, size_t ws_size,
                              hipStream_t stream)
{
    const float* x   = (const float*)d_in[0];
    const float* w1  = (const float*)d_in[1];
    const float* cb1 = (const float*)d_in[2];
    const float* w2  = (const float*)d_in[3];
    const float* cb2 = (const float*)d_in[4];
    const float* w3  = (const float*)d_in[5];
    const float* cb3 = (const float*)d_in[6];
    const float* g1  = (const float*)d_in[7];
    const float* be1 = (const float*)d_in[8];
    const float* g2  = (const float*)d_in[9];
    const float* be2 = (const float*)d_in[10];
    const float* g3  = (const float*)d_in[11];
    const float* be3 = (const float*)d_in[12];
    const float* fcw = (const float*)d_in[13];
    const float* fcb = (const float*)d_in[14];
    const float* pr  = (const float*)d_in[15];

    float* hout = (float*)d_out;                       // [B,C3,T] goes first
    float* oout = hout + (size_t)BATCH * C3 * TLEN;    // then [B,NCLS]

    float* ws = (float*)d_ws;
    size_t o = 0;
    float* buf1   = ws + o; o += (size_t)BATCH * C1 * TLEN;          // 16.8M
    float* buf2   = ws + o; o += (size_t)BATCH * C2 * TLEN;          // 33.6M
    float* Dbuf   = ws + o; o += (size_t)BATCH * SPROT * TLEN;       // 2.1M
    float* Rbuf   = ws + o; o += (size_t)BATCH * (SPROT + 2) * (TLEN + 2);
    float* Abuf   = ws + o; o += (size_t)BATCH * SPROT * TLEN;
    float* pooled = ws + o; o += (size_t)BATCH * C3 * SPROT;
    float* sc     = ws + o; o += 256;
    float* sf     = ws + o; o += 256;
    float* pn     = ws + o; o += 16;

    // ---- layer 1: conv(64->128,k9,p4) + BN + ReLU ----
    {
        dim3 grid(TLEN / 16, C1 / 128, BATCH);         // (64, 1, 128)
        conv_wmma_kernel<9, CIN, C1><<<grid, 256, 0, stream>>>(x, w1, cb1, buf1);
        bn_stats_kernel<<<C1, 256, 0, stream>>>(buf1, g1, be1, sc, sf, C1);
        size_t tot = (size_t)BATCH * C1 * TLEN;
        bn_relu_kernel<<<(int)(tot / 256), 256, 0, stream>>>(buf1, sc, sf, C1, tot);
    }
    // ---- layer 2: conv(128->256,k5,p2) + BN + ReLU ----
    {
        dim3 grid(TLEN / 16, C2 / 128, BATCH);         // (64, 2, 128)
        conv_wmma_kernel<5, C1, C2><<<grid, 256, 0, stream>>>(buf1, w2, cb2, buf2);
        bn_stats_kernel<<<C2, 256, 0, stream>>>(buf2, g2, be2, sc, sf, C2);
        size_t tot = (size_t)BATCH * C2 * TLEN;
        bn_relu_kernel<<<(int)(tot / 256), 256, 0, stream>>>(buf2, sc, sf, C2, tot);
    }
    // ---- layer 3: conv(256->128,k3,p1) + BN + ReLU, straight into d_out ----
    {
        dim3 grid(TLEN / 16, C3 / 128, BATCH);         // (64, 1, 128)
        conv_wmma_kernel<3, C2, C3><<<grid, 256, 0, stream>>>(buf2, w3, cb3, hout);
        bn_stats_kernel<<<C3, 256, 0, stream>>>(hout, g3, be3, sc, sf, C3);
        size_t tot = (size_t)BATCH * C3 * TLEN;
        bn_relu_kernel<<<(int)(tot / 256), 256, 0, stream>>>(hout, sc, sf, C3, tot);
    }
    // ---- alignment ----
    pnorm_kernel<<<1, 16, 0, stream>>>(pr, pn);
    cost_kernel<<<dim3(TLEN / 256, BATCH), 256, 0, stream>>>(hout, pr, pn, Dbuf);
    dtw_kernel<<<BATCH, 32, 0, stream>>>(Dbuf, Rbuf, Abuf);
    arow_norm_kernel<<<BATCH * SPROT, 256, 0, stream>>>(Abuf);
    // ---- pooling + classifier ----
    pool_kernel<<<dim3(C3, BATCH), 256, 0, stream>>>(hout, Abuf, pooled);
    fc_kernel<<<BATCH, 32, 0, stream>>>(pooled, fcw, fcb, oout);
}